// BYOLModel_90778428768362
// MI455X (gfx1250) — compile-verified
//
#include <hip/hip_runtime.h>
#include <hip/hip_bf16.h>
#include <math.h>

#define HW   196
#define CCH  2048
#define BATCH 256
#define HB   128
#define EPSV 1e-12f
#define BNEPS 1e-5f

typedef __attribute__((ext_vector_type(2))) float v2f;
typedef __attribute__((ext_vector_type(8))) float v8f;

// ---------------------------------------------------------------------------
// Kernel 1: pooled[b][c] = mean over 196 pixels of emb_a[b,c,:,:]
// emb_a[b] = view1[b] for b<128, view2[b-128] otherwise.
// One block per b; thread t handles channels t, t+256, ... (8 channels).
// ---------------------------------------------------------------------------
__global__ __launch_bounds__(256) void pool_kernel(const float* __restrict__ v1,
                                                   const float* __restrict__ v2,
                                                   float* __restrict__ pooled) {
  const int b = blockIdx.x;
  const float* src = (b < HB) ? (v1 + (size_t)b * CCH * HW)
                              : (v2 + (size_t)(b - HB) * CCH * HW);
  const int t = threadIdx.x;
  for (int j = 0; j < CCH / 256; ++j) {
    const int c = t + j * 256;
    const float4* row = (const float4*)(src + (size_t)c * HW);
    float s = 0.f;
    for (int q = 0; q < HW / 4; ++q) {
      float4 x = row[q];
      s += x.x + x.y + x.z + x.w;
    }
    pooled[(size_t)b * CCH + c] = s * (1.0f / (float)HW);
  }
}

// ---------------------------------------------------------------------------
// Kernel 2: per-sample attention + weighted channel sum.
//  - normalize partner pooled vector in LDS
//  - pass 1 (pixel-per-lane): dot(emb[:,p], pn) and sum_c emb^2 per pixel
//  - atten = ((dot/max(||pix||,eps)) + 1)/2, normalized by its sum
//  - writes atten_a[b] and atten_b[(b+128)%256] (swap identity)
//  - pass 2 (channel-major, float4): ea[b][c] = sum_p emb[c,p]*atten[p]
// ---------------------------------------------------------------------------
__global__ __launch_bounds__(256) void atten_ea_kernel(const float* __restrict__ v1,
                                                       const float* __restrict__ v2,
                                                       const float* __restrict__ pooled,
                                                       float* __restrict__ ea,
                                                       float* __restrict__ out_atten_a,
                                                       float* __restrict__ out_atten_b) {
  __shared__ float pn[CCH];                   // normalized partner pooled (8 KB)
  __shared__ float red[256];
  __shared__ __align__(16) float att[HW];     // final attention map
  const int b = blockIdx.x;
  const int t = threadIdx.x;
  const int bp = (b + HB) & (BATCH - 1);      // partner batch index
  const float* src = (b < HB) ? (v1 + (size_t)b * CCH * HW)
                              : (v2 + (size_t)(b - HB) * CCH * HW);

  // load partner pooled + sum of squares
  float ss = 0.f;
  for (int j = 0; j < CCH / 256; ++j) {
    const int c = t + j * 256;
    const float x = pooled[(size_t)bp * CCH + c];
    pn[c] = x;
    ss += x * x;
  }
  red[t] = ss;
  __syncthreads();
  for (int off = 128; off > 0; off >>= 1) {
    if (t < off) red[t] += red[t + off];
    __syncthreads();
  }
  const float inv = 1.0f / fmaxf(sqrtf(red[0]), EPSV);
  for (int j = 0; j < CCH / 256; ++j) pn[t + j * 256] *= inv;
  __syncthreads();

  // pass 1: per-pixel cosine numerator and pixel norm
  float a = 0.f;
  if (t < HW) {
    float dot = 0.f, sq = 0.f;
    for (int c = 0; c < CCH; ++c) {
      const float x = src[(size_t)c * HW + t];
      dot = fmaf(x, pn[c], dot);
      sq  = fmaf(x, x, sq);
    }
    const float cosv = dot / fmaxf(sqrtf(sq), EPSV);
    a = (cosv + 1.0f) * 0.5f;                 // nonnegative -> |a| == a
  }
  red[t] = (t < HW) ? a : 0.f;
  __syncthreads();
  for (int off = 128; off > 0; off >>= 1) {
    if (t < off) red[t] += red[t + off];
    __syncthreads();
  }
  const float denom = fmaxf(red[0], EPSV);
  if (t < HW) {
    const float av = a / denom;
    att[t] = av;
    out_atten_a[(size_t)b  * HW + t] = av;
    out_atten_b[(size_t)bp * HW + t] = av;    // atten_b = swapHalves(atten_a)
  }
  __syncthreads();

  // pass 2: ea[b][c] = sum_p emb[c,p] * att[p]
  const float4* at4 = (const float4*)att;
  for (int j = 0; j < CCH / 256; ++j) {
    const int c = t + j * 256;
    const float4* row = (const float4*)(src + (size_t)c * HW);
    float s = 0.f;
    for (int q = 0; q < HW / 4; ++q) {
      const float4 x = row[q];
      const float4 w = at4[q];
      s = fmaf(x.x, w.x, s);
      s = fmaf(x.y, w.y, s);
      s = fmaf(x.z, w.z, s);
      s = fmaf(x.w, w.w, s);
    }
    ea[(size_t)b * CCH + c] = s;
  }
}

// ---------------------------------------------------------------------------
// FP32 WMMA GEMM: C[256 x N] = A[256 x K] * B[K x N] (+ bias), optional
// swapped-halves copy to C2 (proj_b trick). Block = 8 waves; wave w owns rows
// [w*32, w*32+32); block owns a 64-wide column strip, so each weight matrix B
// is streamed from HBM exactly once across the grid.
// A frag (16x4 f32): lanes 0-15 hold M=lane, VGPR0/1 = K pair selected by
// lane-half (K+0,K+1 for lanes<16; K+2,K+3 for lanes>=16) -> one float2 load.
// B frag (4x16 f32): lanes 0-15 hold N=lane, same K-pair-by-lane-half rule.
// ---------------------------------------------------------------------------
__global__ __launch_bounds__(256) void gemm_wmma_kernel(const float* __restrict__ A,
                                                        const float* __restrict__ B,
                                                        const float* __restrict__ bias,
                                                        float* __restrict__ C,
                                                        float* __restrict__ C2,
                                                        int K, int N) {
  const int lane  = threadIdx.x & 31;
  const int wave  = threadIdx.x >> 5;
  const int n0    = blockIdx.x * 64;
  const int mbase = wave * 32;
  const int lrow  = lane & 15;
  const int khalf = lane >> 4;                 // 0 or 1

  const v8f vzero = {0.f, 0.f, 0.f, 0.f, 0.f, 0.f, 0.f, 0.f};
  v8f acc[2][4];
  for (int mt = 0; mt < 2; ++mt)
    for (int nt = 0; nt < 4; ++nt)
      acc[mt][nt] = vzero;

  for (int k = 0; k < K; k += 4) {
    const int ka = k + 2 * khalf;
    const float2 fa0 = *(const float2*)(A + (size_t)(mbase + lrow)      * K + ka);
    const float2 fa1 = *(const float2*)(A + (size_t)(mbase + 16 + lrow) * K + ka);
    v2f a0; a0.x = fa0.x; a0.y = fa0.y;
    v2f a1; a1.x = fa1.x; a1.y = fa1.y;
    for (int nt = 0; nt < 4; ++nt) {
      const int col = n0 + nt * 16 + lrow;
      v2f bb;
      bb.x = B[(size_t)ka       * N + col];
      bb.y = B[(size_t)(ka + 1) * N + col];
      acc[0][nt] = __builtin_amdgcn_wmma_f32_16x16x4_f32(
          false, a0, false, bb, (short)0, acc[0][nt], false, false);
      acc[1][nt] = __builtin_amdgcn_wmma_f32_16x16x4_f32(
          false, a1, false, bb, (short)0, acc[1][nt], false, false);
    }
  }

  // epilogue: C/D layout -> VGPR j holds rows M=j (lanes 0-15) / M=j+8 (16-31)
  for (int mt = 0; mt < 2; ++mt) {
    for (int nt = 0; nt < 4; ++nt) {
      const int col = n0 + nt * 16 + lrow;
      const float bv = bias ? bias[col] : 0.f;
      for (int j = 0; j < 8; ++j) {
        const int row = mbase + mt * 16 + j + 8 * khalf;
        const float v = acc[mt][nt][j] + bv;
        C[(size_t)row * N + col] = v;
        if (C2) C2[(size_t)((row + HB) & (BATCH - 1)) * N + col] = v;
      }
    }
  }
}

// ---------------------------------------------------------------------------
// In-place BatchNorm (over the 256-row batch axis) + ReLU, one thread/column.
// ---------------------------------------------------------------------------
__global__ __launch_bounds__(256) void bn_relu_kernel(float* __restrict__ Y,
                                                      const float* __restrict__ g,
                                                      const float* __restrict__ be,
                                                      int N) {
  const int col = blockIdx.x * blockDim.x + threadIdx.x;
  if (col >= N) return;
  float s = 0.f, s2 = 0.f;
  for (int r = 0; r < BATCH; ++r) {
    const float x = Y[(size_t)r * N + col];
    s += x;
    s2 = fmaf(x, x, s2);
  }
  const float m   = s * (1.0f / (float)BATCH);
  const float var = s2 * (1.0f / (float)BATCH) - m * m;
  const float sc  = g[col] * rsqrtf(var + BNEPS);
  const float sh  = be[col] - m * sc;
  for (int r = 0; r < BATCH; ++r) {
    const float x = Y[(size_t)r * N + col];
    Y[(size_t)r * N + col] = fmaxf(fmaf(x, sc, sh), 0.f);
  }
}

// ---------------------------------------------------------------------------
// Host launcher. Input order: view1, view2, mm, W1, b1, g1, be1, W2, b2,
//                             P1, pb1, pg1, pbe1, P2, pb2.
// Output: [q (65536) | proj_b (65536) | atten_a (50176) | atten_b (50176)].
// Note: b1/pb1 cancel inside BN and are skipped; mm is unused by reference.
// ---------------------------------------------------------------------------
extern "C" void kernel_launch(void* const* d_in, const int* in_sizes, int n_in,
                              void* d_out, int out_size, void* d_ws, size_t ws_size,
                              hipStream_t stream) {
  (void)in_sizes; (void)n_in; (void)out_size; (void)ws_size;
  const float* view1 = (const float*)d_in[0];
  const float* view2 = (const float*)d_in[1];
  const float* W1 = (const float*)d_in[3];
  const float* g1 = (const float*)d_in[5];
  const float* be1 = (const float*)d_in[6];
  const float* W2 = (const float*)d_in[7];
  const float* b2 = (const float*)d_in[8];
  const float* P1 = (const float*)d_in[9];
  const float* pg1 = (const float*)d_in[11];
  const float* pbe1 = (const float*)d_in[12];
  const float* P2 = (const float*)d_in[13];
  const float* pb2 = (const float*)d_in[14];

  float* out = (float*)d_out;
  float* out_q       = out;                 // 256*256
  float* out_projb   = out + 65536;         // 256*256
  float* out_atten_a = out + 131072;        // 256*196
  float* out_atten_b = out + 181248;        // 256*196

  float* ws = (float*)d_ws;
  float* ws_pooled = ws;                    // 256*2048 = 524288 floats
  float* ws_ea     = ws + 524288;           // 256*2048
  float* ws_Y1     = ws + 1048576;          // 256*4096 = 1048576 floats
  float* ws_projA  = ws + 2097152;          // 256*256  = 65536 floats
  float* ws_Y2     = ws;                    // reuse pooled+ea region (dead)

  // 1) pooled means
  pool_kernel<<<BATCH, 256, 0, stream>>>(view1, view2, ws_pooled);
  // 2) attention + ea (also emits atten_a and atten_b outputs)
  atten_ea_kernel<<<BATCH, 256, 0, stream>>>(view1, view2, ws_pooled, ws_ea,
                                             out_atten_a, out_atten_b);
  // 3) Y1 = ea @ W1 (bias cancels in BN)
  gemm_wmma_kernel<<<4096 / 64, 256, 0, stream>>>(ws_ea, W1, nullptr, ws_Y1,
                                                  nullptr, CCH, 4096);
  // 4) BN + ReLU in place
  bn_relu_kernel<<<4096 / 256, 256, 0, stream>>>(ws_Y1, g1, be1, 4096);
  // 5) proj_a = H1 @ W2 + b2 ; proj_b = swapped-halves copy
  gemm_wmma_kernel<<<256 / 64, 256, 0, stream>>>(ws_Y1, W2, b2, ws_projA,
                                                 out_projb, 4096, 256);
  // 6) Y2 = proj_a @ P1 (bias cancels in BN)
  gemm_wmma_kernel<<<4096 / 64, 256, 0, stream>>>(ws_projA, P1, nullptr, ws_Y2,
                                                  nullptr, 256, 4096);
  // 7) BN + ReLU in place
  bn_relu_kernel<<<4096 / 256, 256, 0, stream>>>(ws_Y2, pg1, pbe1, 4096);
  // 8) q = H2 @ P2 + pb2
  gemm_wmma_kernel<<<256 / 64, 256, 0, stream>>>(ws_Y2, P2, pb2, out_q,
                                                 nullptr, 4096, 256);
}